// GNN_31851477467287
// MI455X (gfx1250) — compile-verified
//
#include <hip/hip_runtime.h>

// ---------------------------------------------------------------------------
// 2-layer GCN for gfx1250 (MI455X).
//   features[N,128] f32, W1[128,64], b1[64], W2[64,2], b2[2],
//   src[E] i64, dst[E] i64  ->  out[N,2] f32
// Layer-1 projection uses V_WMMA_F32_16X16X4_F32 (exact f32, matches ref).
// Edge phase (the bandwidth-bound part) = wave-per-edge gather + f32 atomics.
// ---------------------------------------------------------------------------

typedef __attribute__((ext_vector_type(2))) float v2f;
typedef __attribute__((ext_vector_type(8))) float v8f;

#define IN_FEATS 128
#define HIDDEN   64
#define NCLS     2

// ----------------------------- utility kernels -----------------------------

__global__ void gnn_zero_f32(float* __restrict__ p, long long n) {
  long long i = (long long)blockIdx.x * blockDim.x + threadIdx.x;
  if (i < n) p[i] = 0.0f;
}

__global__ void gnn_degree(const long long* __restrict__ src,
                           const long long* __restrict__ dst,
                           float* __restrict__ deg_src,
                           float* __restrict__ deg_dst, long long E) {
  long long e = (long long)blockIdx.x * blockDim.x + threadIdx.x;
  if (e < E) {
    atomicAdd(&deg_src[src[e]], 1.0f);
    atomicAdd(&deg_dst[dst[e]], 1.0f);
  }
}

// deg -> deg^{-1/2} with clamp(deg,1) in place, for both buffers
__global__ void gnn_invsqrt(float* __restrict__ a, float* __restrict__ b,
                            long long n) {
  long long i = (long long)blockIdx.x * blockDim.x + threadIdx.x;
  if (i < n) {
    a[i] = rsqrtf(fmaxf(a[i], 1.0f));
    b[i] = rsqrtf(fmaxf(b[i], 1.0f));
  }
}

// ------------------- layer 1 projection: f32 WMMA GEMM ---------------------
// H1[m, n] = sum_k (X[m,k] * norm_src[m]) * W1[k,n]
// Block = 128 threads = 4 waves; wave w -> col tile n0 = 16*w (HIDDEN=64).
// Each wave: 16x16 f32 tile, K-loop of 32 x V_WMMA_F32_16X16X4_F32.
__global__ void gnn_gemm1_wmma(const float* __restrict__ X,
                               const float* __restrict__ W,
                               const float* __restrict__ norm_src,
                               float* __restrict__ H, int Nn) {
  const int lane = threadIdx.x & 31;
  const int wave = threadIdx.x >> 5;   // 0..3 -> column tile
  const int half = lane >> 4;          // 0 or 1 (lane group)
  const int ml   = lane & 15;

  const int m0 = blockIdx.x * 16;
  const int n0 = wave * 16;

  int row = m0 + ml;
  if (row >= Nn) row = Nn - 1;         // clamp loads; stores guarded below
  const float  s  = norm_src[row];
  const float* xr = X + (long long)row * IN_FEATS;

  v8f acc = {};
  #pragma unroll
  for (int k0 = 0; k0 < IN_FEATS; k0 += 4) {
    // A 16x4 f32 layout: VGPR0 = K = k0+2*half, VGPR1 = K = k0+2*half+1
    const int ka = k0 + 2 * half;
    v2f a;
    a.x = xr[ka]     * s;
    a.y = xr[ka + 1] * s;
    // B 4x16 f32: row striped across lanes within a VGPR, mirrored halves
    v2f b;
    b.x = W[(long long)ka       * HIDDEN + n0 + ml];
    b.y = W[(long long)(ka + 1) * HIDDEN + n0 + ml];
    acc = __builtin_amdgcn_wmma_f32_16x16x4_f32(
        /*neg_a=*/false, a, /*neg_b=*/false, b,
        /*c_mod=*/(short)0, acc, /*reuse_a=*/false, /*reuse_b=*/false);
  }

  // C/D 16x16 f32 layout: VGPR i holds row m0 + i + 8*half, col n0 + ml
  #pragma unroll
  for (int i = 0; i < 8; ++i) {
    const int r = m0 + i + 8 * half;
    if (r < Nn) H[(long long)r * HIDDEN + n0 + ml] = acc[i];
  }
}

// ------------------- layer 1 edge aggregation (wave/edge) ------------------
// lane l moves features 2l and 2l+1 (float2 gather + 2 f32 atomics)
__global__ void gnn_agg1(const long long* __restrict__ src,
                         const long long* __restrict__ dst,
                         const float* __restrict__ H,
                         float* __restrict__ A, long long E) {
  long long gid = (long long)blockIdx.x * blockDim.x + threadIdx.x;
  long long e = gid >> 5;
  if (e >= E) return;
  const int lane = (int)(gid & 31);
  const long long s = src[e];
  const long long d = dst[e];
  const float2 v = *(const float2*)(H + s * HIDDEN + 2 * lane);
  atomicAdd(&A[d * HIDDEN + 2 * lane],     v.x);
  atomicAdd(&A[d * HIDDEN + 2 * lane + 1], v.y);
}

// relu(A * norm_dst + b1) in place
__global__ void gnn_fin1(float* __restrict__ A,
                         const float* __restrict__ norm_dst,
                         const float* __restrict__ b1, long long total) {
  long long i = (long long)blockIdx.x * blockDim.x + threadIdx.x;
  if (i < total) {
    const long long node = i >> 6;      // /HIDDEN
    const int f = (int)(i & 63);
    const float x = A[i] * norm_dst[node] + b1[f];
    A[i] = x > 0.0f ? x : 0.0f;
  }
}

// ------------------------- layer 2 (out dim = 2) ---------------------------
// H2[n,c] = norm_src[n] * sum_k Hr[n,k] * W2[k,c]   (N_out=2 -> VALU dot)
__global__ void gnn_proj2(const float* __restrict__ Hr,
                          const float* __restrict__ norm_src,
                          const float* __restrict__ W2,
                          float* __restrict__ H2, long long Nn) {
  long long i = (long long)blockIdx.x * blockDim.x + threadIdx.x;
  if (i < Nn * NCLS) {
    const long long n = i >> 1;
    const int c = (int)(i & 1);
    const float* hr = Hr + n * HIDDEN;
    float acc = 0.0f;
    #pragma unroll
    for (int k = 0; k < HIDDEN; ++k) acc += hr[k] * W2[k * NCLS + c];
    H2[i] = acc * norm_src[n];
  }
}

__global__ void gnn_agg2(const long long* __restrict__ src,
                         const long long* __restrict__ dst,
                         const float* __restrict__ H2,
                         float* __restrict__ out, long long E) {
  long long gid = (long long)blockIdx.x * blockDim.x + threadIdx.x;
  long long e = gid >> 1;
  if (e >= E) return;
  const int c = (int)(gid & 1);
  atomicAdd(&out[dst[e] * NCLS + c], H2[src[e] * NCLS + c]);
}

__global__ void gnn_fin2(float* __restrict__ out,
                         const float* __restrict__ norm_dst,
                         const float* __restrict__ b2, long long Nn) {
  long long i = (long long)blockIdx.x * blockDim.x + threadIdx.x;
  if (i < Nn * NCLS) {
    out[i] = out[i] * norm_dst[i >> 1] + b2[i & 1];
  }
}

// ------------------------------- launcher ----------------------------------

static inline int nblk(long long n, int t) { return (int)((n + t - 1) / t); }

extern "C" void kernel_launch(void* const* d_in, const int* in_sizes, int n_in,
                              void* d_out, int out_size, void* d_ws, size_t ws_size,
                              hipStream_t stream) {
  const float*     feat = (const float*)d_in[0];       // [N,128]
  const float*     W1   = (const float*)d_in[1];       // [128,64]
  const float*     b1   = (const float*)d_in[2];       // [64]
  const float*     W2   = (const float*)d_in[3];       // [64,2]
  const float*     b2   = (const float*)d_in[4];       // [2]
  const long long* src  = (const long long*)d_in[5];   // [E] int64
  const long long* dst  = (const long long*)d_in[6];   // [E] int64

  const long long Nn = (long long)in_sizes[0] / IN_FEATS;
  const long long E  = (long long)in_sizes[5];

  // workspace layout (floats):
  //   [0, N)        norm_src (deg accumulator, then rsqrt)
  //   [N, 2N)       norm_dst
  //   [2N, 66N)     H1  (layer-1 projection; later reused as H2[2N])
  //   [66N, 130N)   AGG1 (layer-1 aggregation / hidden activations)
  float* ws       = (float*)d_ws;
  float* norm_src = ws;
  float* norm_dst = ws + Nn;
  float* H1       = ws + 2 * Nn;
  float* AGG1     = ws + 2 * Nn + (long long)HIDDEN * Nn;
  float* out      = (float*)d_out;

  // --- zero accumulators (every launch: deterministic, ws is not re-poisoned)
  gnn_zero_f32<<<nblk(2 * Nn, 256), 256, 0, stream>>>(norm_src, 2 * Nn);
  gnn_zero_f32<<<nblk((long long)HIDDEN * Nn, 256), 256, 0, stream>>>(
      AGG1, (long long)HIDDEN * Nn);
  gnn_zero_f32<<<nblk(NCLS * Nn, 256), 256, 0, stream>>>(out, NCLS * Nn);

  // --- degrees -> symmetric norms
  gnn_degree<<<nblk(E, 256), 256, 0, stream>>>(src, dst, norm_src, norm_dst, E);
  gnn_invsqrt<<<nblk(Nn, 256), 256, 0, stream>>>(norm_src, norm_dst, Nn);

  // --- layer 1: f32 WMMA projection, edge aggregation, relu
  gnn_gemm1_wmma<<<nblk(Nn, 16), 128, 0, stream>>>(feat, W1, norm_src, H1, (int)Nn);
  gnn_agg1<<<nblk(E * 32, 256), 256, 0, stream>>>(src, dst, H1, AGG1, E);
  gnn_fin1<<<nblk((long long)HIDDEN * Nn, 256), 256, 0, stream>>>(
      AGG1, norm_dst, b1, (long long)HIDDEN * Nn);

  // --- layer 2: tiny projection (reuse H1 region for H2), aggregate into out
  gnn_proj2<<<nblk(NCLS * Nn, 256), 256, 0, stream>>>(AGG1, norm_src, W2, H1, Nn);
  gnn_agg2<<<nblk(E * NCLS, 256), 256, 0, stream>>>(src, dst, H1, out, E);
  gnn_fin2<<<nblk(NCLS * Nn, 256), 256, 0, stream>>>(out, norm_dst, b2, Nn);
}